// VectorQuantizer_61710090109133
// MI455X (gfx1250) — compile-verified
//
#include <hip/hip_runtime.h>
#include <hip/hip_bf16.h>

typedef float v2f __attribute__((ext_vector_type(2)));
typedef float v8f __attribute__((ext_vector_type(8)));

#define K_CODES 8192
#define CDIM 32
#define NROWS 32768                       // 8*16*16*16
#define ROWS_PER_WAVE 32
#define WAVES_PER_BLOCK 4
#define THREADS_PER_BLOCK (WAVES_PER_BLOCK * 32)
#define ROWS_PER_BLOCK (ROWS_PER_WAVE * WAVES_PER_BLOCK)
#define NUM_BLOCKS (NROWS / ROWS_PER_BLOCK)     // 256
#define NUM_KTILES (K_CODES / 16)               // 512

// Output layout (flat, return order): quantized_st, vq_loss, commitment_loss, idx
#define OUT_Q 0
#define OUT_VQ (NROWS * CDIM)            // 1048576
#define OUT_COMMIT (OUT_VQ + 1)
#define OUT_IDX (OUT_COMMIT + 1)         // 1048578

// Workspace layout: [0, 8192) = ||e_n||^2 per code; [8192, 8448) = block loss partials
#define WS_ENORM 0
#define WS_LOSS K_CODES

// ---- Pass 1: precompute ||e_n||^2 for all 8192 codes (runs once, ~1 MB read) ----
__global__ __launch_bounds__(256)
void vq_enorm(const float* __restrict__ emb, float* __restrict__ ws) {
  const int n = blockIdx.x * 256 + threadIdx.x;   // 32 blocks x 256 = 8192
  const float* e = emb + (size_t)n * CDIM;
  float s = 0.f;
#pragma unroll
  for (int c = 0; c < CDIM; c += 4) {
    const float4 v = *(const float4*)(e + c);
    s += v.x * v.x + v.y * v.y + v.z * v.z + v.w * v.w;
  }
  ws[WS_ENORM + n] = s;
}

// ---- Pass 2: fused GEMM + argmin + gather + loss partials ----
__global__ __launch_bounds__(THREADS_PER_BLOCK)
void vq_main(const float* __restrict__ z, const float* __restrict__ emb,
             float* __restrict__ out, float* __restrict__ ws) {
  __shared__ int   s_idx[WAVES_PER_BLOCK * ROWS_PER_WAVE];
  __shared__ float s_loss[WAVES_PER_BLOCK];

  const int lane = threadIdx.x & 31;
  const int wave = threadIdx.x >> 5;
  const int half = lane >> 4;      // which 16-lane half of the wave
  const int col  = lane & 15;      // N index within a 16-wide tile

  const int row_base = (blockIdx.x * WAVES_PER_BLOCK + wave) * ROWS_PER_WAVE;
  const float* __restrict__ enorm_buf = ws + WS_ENORM;

  // ---- A fragments for two 16x32 row tiles (32-bit A layout: lane = M,
  // VGPR pair j covers K = {4j+2h, 4j+2h+1}) ----
  v2f a0[8], a1[8];
  {
    const float* z0 = z + (size_t)(row_base + col) * CDIM + 2 * half;
    const float* z1 = z + (size_t)(row_base + 16 + col) * CDIM + 2 * half;
#pragma unroll
    for (int j = 0; j < 8; ++j) {
      a0[j] = *(const v2f*)(z0 + 4 * j);
      a1[j] = *(const v2f*)(z1 + 4 * j);
    }
  }

  float best0[8], best1[8];
  int   idx0[8], idx1[8];
#pragma unroll
  for (int r = 0; r < 8; ++r) {
    best0[r] = __builtin_inff(); best1[r] = __builtin_inff();
    idx0[r] = 0; idx1[r] = 0;
  }

  // B fragment base for column n = kt*16 + col (same float2 layout as A).
  const float* ebase = emb + (size_t)col * CDIM + 2 * half;

  // Two statically-named B buffers -> all register indices are compile-time
  // constants (no v_movrel traffic).
  v2f bA[8], bB[8];
  float enA, enB;

  // One 16x16 k-tile: 16 chained f32 WMMAs + fused argmin epilogue.
  auto process = [&](const v2f (&b)[8], float en, int kt) {
    v8f acc0 = {0.f, 0.f, 0.f, 0.f, 0.f, 0.f, 0.f, 0.f};
    v8f acc1 = {0.f, 0.f, 0.f, 0.f, 0.f, 0.f, 0.f, 0.f};
#pragma unroll
    for (int j = 0; j < 8; ++j) {
      acc0 = __builtin_amdgcn_wmma_f32_16x16x4_f32(
          false, a0[j], false, b[j], (short)0, acc0, false, false);
      acc1 = __builtin_amdgcn_wmma_f32_16x16x4_f32(
          false, a1[j], false, b[j], (short)0, acc1, false, false);
    }
    const int n = kt * 16 + col;
#pragma unroll
    for (int r = 0; r < 8; ++r) {
      const float s0 = fmaf(acc0[r], -2.0f, en);   // ||e||^2 - 2*z.e
      if (s0 < best0[r]) { best0[r] = s0; idx0[r] = n; }
      const float s1 = fmaf(acc1[r], -2.0f, en);
      if (s1 < best1[r]) { best1[r] = s1; idx1[r] = n; }
    }
  };

  // Preload tile 0 into buffer A.
#pragma unroll
  for (int j = 0; j < 8; ++j) bA[j] = *(const v2f*)(ebase + 4 * j);
  enA = enorm_buf[col];

  for (int kt = 0; kt < NUM_KTILES; kt += 2) {
    // Prefetch tile kt+1 into B while computing tile kt from A.
    {
      const float* e = ebase + (size_t)(kt + 1) * 16 * CDIM;
#pragma unroll
      for (int j = 0; j < 8; ++j) bB[j] = *(const v2f*)(e + 4 * j);
      enB = enorm_buf[(kt + 1) * 16 + col];
    }
    process(bA, enA, kt);

    // Prefetch tile kt+2 into A while computing tile kt+1 from B.
    if (kt + 2 < NUM_KTILES) {
      const float* e = ebase + (size_t)(kt + 2) * 16 * CDIM;
#pragma unroll
      for (int j = 0; j < 8; ++j) bA[j] = *(const v2f*)(e + 4 * j);
      enA = enorm_buf[(kt + 2) * 16 + col];
    }
    process(bB, enB, kt + 1);
  }

  // ---- Argmin butterfly across the 16 lanes of each half; lowest-index
  // tie-break matches jnp.argmin first-occurrence semantics. ----
#pragma unroll
  for (int r = 0; r < 8; ++r) {
#pragma unroll
    for (int m = 8; m >= 1; m >>= 1) {
      float ob = __shfl_xor(best0[r], m, 32);
      int   oi = __shfl_xor(idx0[r], m, 32);
      if (ob < best0[r] || (ob == best0[r] && oi < idx0[r])) { best0[r] = ob; idx0[r] = oi; }
      ob = __shfl_xor(best1[r], m, 32);
      oi = __shfl_xor(idx1[r], m, 32);
      if (ob < best1[r] || (ob == best1[r] && oi < idx1[r])) { best1[r] = ob; idx1[r] = oi; }
    }
  }

  // C/D layout: VGPR r, half h -> local row r + 8h (tile0), 16 + r + 8h (tile1).
  if (col == 0) {
#pragma unroll
    for (int r = 0; r < 8; ++r) {
      s_idx[wave * 32 + 8 * half + r]      = idx0[r];
      s_idx[wave * 32 + 16 + 8 * half + r] = idx1[r];
    }
  }
  __syncthreads();

  // ---- Gather emb[idx], write quantized_st (== quantized numerically),
  // idx as float, and accumulate squared-diff loss. One row per lane. ----
  const int myrow = row_base + lane;
  const int myidx = s_idx[wave * 32 + lane];
  const float* esrc = emb + (size_t)myidx * CDIM;
  const float* zsrc = z + (size_t)myrow * CDIM;
  float* qdst = out + OUT_Q + (size_t)myrow * CDIM;

  float lsum = 0.f;
#pragma unroll
  for (int c = 0; c < CDIM; c += 4) {
    const float4 e  = *(const float4*)(esrc + c);
    const float4 zz = *(const float4*)(zsrc + c);
    *(float4*)(qdst + c) = e;
    const float dx = zz.x - e.x, dy = zz.y - e.y;
    const float dz2 = zz.z - e.z, dw = zz.w - e.w;
    lsum += dx * dx + dy * dy + dz2 * dz2 + dw * dw;
  }
  out[OUT_IDX + myrow] = (float)myidx;

  // Deterministic block-level loss partial: wave shuffle -> LDS -> thread 0.
#pragma unroll
  for (int m = 16; m >= 1; m >>= 1) lsum += __shfl_xor(lsum, m, 32);
  if (lane == 0) s_loss[wave] = lsum;
  __syncthreads();
  if (threadIdx.x == 0) {
    float t = 0.f;
    for (int w = 0; w < WAVES_PER_BLOCK; ++w) t += s_loss[w];
    ws[WS_LOSS + blockIdx.x] = t;
  }
}

// ---- Pass 3: deterministic scalar reduction of the loss ----
__global__ void vq_finalize(const float* __restrict__ ws, float* __restrict__ out) {
  if (threadIdx.x == 0 && blockIdx.x == 0) {
    float t = 0.f;
    for (int i = 0; i < NUM_BLOCKS; ++i) t += ws[WS_LOSS + i];  // fixed order
    const float mean = t / (float)(NROWS * CDIM);
    out[OUT_VQ] = mean;       // vq_loss
    out[OUT_COMMIT] = mean;   // commitment_loss (numerically identical)
  }
}

extern "C" void kernel_launch(void* const* d_in, const int* in_sizes, int n_in,
                              void* d_out, int out_size, void* d_ws, size_t ws_size,
                              hipStream_t stream) {
  const float* z   = (const float*)d_in[0];
  const float* emb = (const float*)d_in[1];
  float* out = (float*)d_out;
  float* ws  = (float*)d_ws;   // needs (8192 + 256) floats = 33 KB scratch

  hipLaunchKernelGGL(vq_enorm, dim3(K_CODES / 256), dim3(256), 0, stream, emb, ws);
  hipLaunchKernelGGL(vq_main, dim3(NUM_BLOCKS), dim3(THREADS_PER_BLOCK), 0, stream,
                     z, emb, out, ws);
  hipLaunchKernelGGL(vq_finalize, dim3(1), dim3(32), 0, stream, ws, out);
}